// SpatialGrid4D_21234318312199
// MI455X (gfx1250) — compile-verified
//
#include <hip/hip_runtime.h>
#include <stdint.h>

// -------- CDNA5 async global->LDS path (probe via __has_builtin) ----------
#ifndef __has_builtin
#define __has_builtin(x) 0
#endif

#if __has_builtin(__builtin_amdgcn_global_load_async_to_lds_b128)
#define ASYNC_LDS 1
#else
#define ASYNC_LDS 0
#endif

// Builtin parameter type (from clang diagnostic): pointer to 16-byte int
// vector; global source is AS1, LDS destination is AS3.
typedef int v4i __attribute__((__vector_size__(4 * sizeof(int))));
typedef __attribute__((address_space(1))) v4i* g_v4i_ptr;   // global
typedef __attribute__((address_space(3))) v4i* l_v4i_ptr;   // LDS

// Native float vector for __builtin_nontemporal_store (HIP float4 is a
// struct and is rejected by the builtin).
typedef float v4f __attribute__((__vector_size__(4 * sizeof(float))));

#define BLK 256
#define QD 32
#define LAT 8
// strides in floats for grid[Q][W][V][U][L]
#define SU  (LAT)                 // 8
#define SV  (QD * LAT)            // 256
#define SW  (QD * QD * LAT)       // 8192
#define SQ  (QD * QD * QD * LAT)  // 262144

__device__ __forceinline__ float4 lerp4(float4 x0, float4 x1, float t, float omt) {
    float4 r;
    r.x = fmaf(t, x1.x, omt * x0.x);
    r.y = fmaf(t, x1.y, omt * x0.y);
    r.z = fmaf(t, x1.z, omt * x0.z);
    r.w = fmaf(t, x1.w, omt * x0.w);
    return r;
}

__global__ __launch_bounds__(BLK) void spatial_grid4d_kernel(
    const float* __restrict__ uvwq,
    const float* __restrict__ grid,
    float* __restrict__ out,
    int n)
{
    const int t   = threadIdx.x;
    const int gid = blockIdx.x * BLK + t;
    if (gid >= n) return;

#if ASYNC_LDS
    __shared__ float s_coord[BLK * 4];
    {
        const float* src = uvwq + (size_t)gid * 4;
        float*       dst = &s_coord[t * 4];
        // per-lane 16B async DMA: global -> LDS, tracked by ASYNCcnt
        __builtin_amdgcn_global_load_async_to_lds_b128(
            (g_v4i_ptr)(uintptr_t)src,
            (l_v4i_ptr)(uint32_t)(uintptr_t)dst,
            /*offset=*/0, /*cpol=*/0);
    }
    // wave-local data: wait on our own ASYNCcnt, no workgroup barrier needed
    asm volatile("s_wait_asynccnt 0" ::: "memory");
    float4 c = *reinterpret_cast<const float4*>(&s_coord[t * 4]);
#else
    float4 c = *reinterpret_cast<const float4*>(uvwq + (size_t)gid * 4);
#endif

    // idxf = coord * (dim-1); idx = trunc; frac = idxf - floor(idxf)
    const float S = 31.0f;
    float fu = c.x * S, fv = c.y * S, fw = c.z * S, fq = c.w * S;
    int iu = (int)fu, iv = (int)fv, iw = (int)fw, iq = (int)fq;
    float a = fu - (float)iu;   // u frac
    float b = fv - (float)iv;   // v frac
    float g = fw - (float)iw;   // w frac
    float d = fq - (float)iq;   // q frac

    // flat base index (fits in int32: max 32^4*8 = 8,388,608)
    int base = ((iq * QD + iw) * QD + iv) * QD * LAT + iu * LAT;
    const float* gp = grid + base;

    const float wv_[2] = {1.0f - b, b};
    const float ww_[2] = {1.0f - g, g};
    const float wq_[2] = {1.0f - d, d};
    const float oma = 1.0f - a;

    float4 accLo = make_float4(0.f, 0.f, 0.f, 0.f);
    float4 accHi = make_float4(0.f, 0.f, 0.f, 0.f);

#pragma unroll
    for (int oq = 0; oq < 2; ++oq) {
#pragma unroll
        for (int ow = 0; ow < 2; ++ow) {
#pragma unroll
            for (int ov = 0; ov < 2; ++ov) {
                // 64 contiguous bytes: latent vectors at u and u+1
                const float4* p = reinterpret_cast<const float4*>(
                    gp + oq * SQ + ow * SW + ov * SV);
                float4 c0 = p[0];  // u,   lat 0-3
                float4 c1 = p[1];  // u,   lat 4-7
                float4 c2 = p[2];  // u+1, lat 0-3
                float4 c3 = p[3];  // u+1, lat 4-7

                float w = wq_[oq] * ww_[ow] * wv_[ov];
                float4 xlo = lerp4(c0, c2, a, oma);
                float4 xhi = lerp4(c1, c3, a, oma);

                accLo.x = fmaf(w, xlo.x, accLo.x);
                accLo.y = fmaf(w, xlo.y, accLo.y);
                accLo.z = fmaf(w, xlo.z, accLo.z);
                accLo.w = fmaf(w, xlo.w, accLo.w);
                accHi.x = fmaf(w, xhi.x, accHi.x);
                accHi.y = fmaf(w, xhi.y, accHi.y);
                accHi.z = fmaf(w, xhi.z, accHi.z);
                accHi.w = fmaf(w, xhi.w, accHi.w);
            }
        }
    }

    // Streaming 128 MB write-once output: non-temporal stores so it does not
    // evict the hot 32 MB grid working set from the 192 MB L2.
    v4f lo = { accLo.x, accLo.y, accLo.z, accLo.w };
    v4f hi = { accHi.x, accHi.y, accHi.z, accHi.w };
    v4f* o = reinterpret_cast<v4f*>(out + (size_t)gid * LAT);
    __builtin_nontemporal_store(lo, &o[0]);   // global_store_b128 th:NT
    __builtin_nontemporal_store(hi, &o[1]);   // global_store_b128 th:NT
}

extern "C" void kernel_launch(void* const* d_in, const int* in_sizes, int n_in,
                              void* d_out, int out_size, void* d_ws, size_t ws_size,
                              hipStream_t stream) {
    (void)n_in; (void)out_size; (void)d_ws; (void)ws_size;
    const float* uvwq = (const float*)d_in[0];
    const float* grid = (const float*)d_in[1];
    float* out        = (float*)d_out;
    int n = in_sizes[0] / 4;   // number of points (N)
    int blocks = (n + BLK - 1) / BLK;
    spatial_grid4d_kernel<<<blocks, BLK, 0, stream>>>(uvwq, grid, out, n);
}